// ConditionalFeaturesUpsample_38594576122336
// MI455X (gfx1250) — compile-verified
//
#include <hip/hip_runtime.h>

typedef __attribute__((ext_vector_type(2))) float v2f;
typedef __attribute__((ext_vector_type(8))) float v8f;
typedef __attribute__((ext_vector_type(4))) float f4;

#define C_IN   80          // channels
#define L_IN   64          // input length
#define T_UP   256         // upsampled length (L_IN * 4)
#define OUT_CH 3072        // 2*BLOCK_HIDDEN*NUM_LAYERS
#define N_LAYERS 24
#define HDIM   128         // 2*BLOCK_HIDDEN
#define REPEAT 75
#define T_FULL (T_UP * REPEAT)   // 19200

// ---------------------------------------------------------------------------
// Stage A: up[c, 4l+k] = sum_c' lf[l,c'] * wt[c',c,k] + bt[c]
//   up: [80, 256] row-major, written to workspace. Tiny (80 KB), VALU is fine.
// ---------------------------------------------------------------------------
__global__ void __launch_bounds__(T_UP)
upsample_kernel(const float* __restrict__ lf, const float* __restrict__ wt,
                const float* __restrict__ bt, float* __restrict__ up) {
    const int c = blockIdx.x;          // 0..79
    const int t = threadIdx.x;         // 0..255
    const int l = t >> 2;
    const int k = t & 3;
    float acc = bt[c];
    const float* lfrow = lf + l * C_IN;       // lf is [L, C]
#pragma unroll 8
    for (int cp = 0; cp < C_IN; ++cp)
        acc += lfrow[cp] * wt[(cp * C_IN + c) * 4 + k];
    up[c * T_UP + t] = acc;
}

// ---------------------------------------------------------------------------
// Stage B: y[o,t] = sum_c w1[o,c] * up[c,t] + b1[o]
//   [3072,80] @ [80,256] via V_WMMA_F32_16X16X4_F32. One wave per 16x16 tile.
//   3072 tiles total; 8 waves / block -> 384 blocks.
//
//   f32 A (16x4) layout: lanes 0-15: M=lane, VGPR0=K0, VGPR1=K1
//                        lanes 16-31: M=lane-16, VGPR0=K2, VGPR1=K3
//   f32 B (4x16) layout (mirrored): lanes 0-15: N=lane, K0/K1; lanes 16-31: K2/K3
//   C/D layout: VGPR r -> row (r + 8*(lane>=16)), col (lane&15)
// ---------------------------------------------------------------------------
__global__ void __launch_bounds__(256)
gemm_wmma_kernel(const float* __restrict__ w1, const float* __restrict__ b1,
                 const float* __restrict__ up, float* __restrict__ y) {
    const int lane = threadIdx.x & 31;
    const int wave = threadIdx.x >> 5;
    const int tile = blockIdx.x * 8 + wave;   // 0..3071
    const int mt = tile >> 4;                 // 0..191
    const int nt = tile & 15;                 // 0..15
    const int m0 = mt * 16;
    const int n0 = nt * 16;
    const int half = lane >> 4;               // 0 or 1
    const int lid  = lane & 15;
    const int rowA = m0 + lid;                // A row this lane holds
    const int colB = n0 + lid;                // B/N col this lane holds

    v8f acc = {};
#pragma unroll
    for (int k0 = 0; k0 < C_IN; k0 += 4) {
        const int ka = k0 + half * 2;         // K pair owned by this lane half
        v2f a, b;
        a.x = w1[rowA * C_IN + ka];
        a.y = w1[rowA * C_IN + ka + 1];
        b.x = up[ka * T_UP + colB];
        b.y = up[(ka + 1) * T_UP + colB];
        acc = __builtin_amdgcn_wmma_f32_16x16x4_f32(
            /*neg_a=*/false, a, /*neg_b=*/false, b,
            /*c_mod=*/(short)0, acc, /*reuse_a=*/false, /*reuse_b=*/false);
    }

#pragma unroll
    for (int r = 0; r < 8; ++r) {
        const int row = m0 + r + half * 8;    // output channel o
        y[row * T_UP + colB] = acc[r] + b1[row];
    }
}

// ---------------------------------------------------------------------------
// Stage C: out[h][layer][T] = y[layer*128 + h][T % 256]
//   out flat: 128 * 24 * 19200 floats = 236 MB -> pure write-bandwidth bound.
//   float4 per thread; NT stores so the 236MB stream doesn't evict hot y (3MB).
//   base is 4-aligned and 19200 % 4 == 0, so the 4 T's share one 256-period
//   and map to a contiguous float4 in y.
// ---------------------------------------------------------------------------
__global__ void __launch_bounds__(256)
tile_permute_kernel(const float* __restrict__ y, float* __restrict__ out) {
    const long long idx  = (long long)blockIdx.x * blockDim.x + threadIdx.x;
    const long long base = idx * 4;
    const int h     = (int)(base / (long long)(N_LAYERS * T_FULL));
    const int rem   = (int)(base % (long long)(N_LAYERS * T_FULL));
    const int layer = rem / T_FULL;
    const int T     = rem % T_FULL;
    const int t     = T & (T_UP - 1);        // T mod 256
    const int o     = layer * HDIM + h;
    const f4 v = *(const f4*)(y + o * T_UP + t);
    __builtin_nontemporal_store(v, (f4*)(out + base));
}

// ---------------------------------------------------------------------------
extern "C" void kernel_launch(void* const* d_in, const int* in_sizes, int n_in,
                              void* d_out, int out_size, void* d_ws, size_t ws_size,
                              hipStream_t stream) {
    const float* lf = (const float*)d_in[0];   // [1,64,80]
    const float* wt = (const float*)d_in[1];   // [80,80,4]
    const float* bt = (const float*)d_in[2];   // [80]
    const float* w1 = (const float*)d_in[3];   // [3072,80]
    const float* b1 = (const float*)d_in[4];   // [3072]
    float* out = (float*)d_out;                // [128,24,19200]

    float* up = (float*)d_ws;                  // 80*256 floats (padded slot 128*256)
    float* y  = (float*)d_ws + 128 * 256;      // 3072*256 floats

    upsample_kernel<<<C_IN, T_UP, 0, stream>>>(lf, wt, bt, up);
    gemm_wmma_kernel<<<384, 256, 0, stream>>>(w1, b1, up, y);

    const long long total_f4 = (long long)HDIM * N_LAYERS * T_FULL / 4;  // 14,745,600
    const int blocks = (int)(total_f4 / 256);                            // 57,600
    tile_permute_kernel<<<blocks, 256, 0, stream>>>(y, out);
}